// LinkConv_65163243815763
// MI455X (gfx1250) — compile-verified
//
#include <hip/hip_runtime.h>

#define N_NODES 50000
#define H_DIM   64
#define E_EDGES 800000

typedef __attribute__((ext_vector_type(2))) float v2f;
typedef __attribute__((ext_vector_type(8))) float v8f;

// ---------------------------------------------------------------------------
// Zero the aggregation workspace (float4-vectorized).
// ---------------------------------------------------------------------------
__global__ __launch_bounds__(256) void zero_kernel(float4* __restrict__ p, int n4) {
  int i = blockIdx.x * 256 + threadIdx.x;
  if (i < n4) p[i] = float4{0.f, 0.f, 0.f, 0.f};
}

// ---------------------------------------------------------------------------
// Edge aggregation: out[dst[e]][:] += feat[e][:] * h[src[e]][:]
// 32 threads per edge, 2 features per thread (float2 loads), f32 HW atomics.
// This is the bandwidth-bound phase (~410 MB feat stream per layer).
// ---------------------------------------------------------------------------
__global__ __launch_bounds__(256) void agg_kernel(const float* __restrict__ h,
                                                  const float* __restrict__ feat,
                                                  const int*   __restrict__ src,
                                                  const int*   __restrict__ dst,
                                                  float*       __restrict__ out) {
  int idx = blockIdx.x * 256 + threadIdx.x;   // E * 32 threads total
  int e   = idx >> 5;
  if (e >= E_EDGES) return;
  int f2  = (idx & 31) << 1;                  // feature pair base

  int s = src[e];
  int d = dst[e];
  const float2 fv = *(const float2*)(feat + (long)e * H_DIM + f2);
  const float2 hv = *(const float2*)(h    + (long)s * H_DIM + f2);

  float* o = out + (long)d * H_DIM + f2;
  __hip_atomic_fetch_add(o + 0, fv.x * hv.x, __ATOMIC_RELAXED, __HIP_MEMORY_SCOPE_AGENT);
  __hip_atomic_fetch_add(o + 1, fv.y * hv.y, __ATOMIC_RELAXED, __HIP_MEMORY_SCOPE_AGENT);
}

// ---------------------------------------------------------------------------
// Fused GEMM + bias + ReLU + residual using V_WMMA_F32_16X16X4_F32.
//   h[node][col] += relu( sum_k concat(a1,a2)[node][k] * W[k][col] + bias[col] )
// One wave = one 16x16 output tile; K=128 -> 32 chained WMMA ops.
//
// A (16x4 f32) layout: lane L holds row M=L%16, K-pair {2*(L>>4), 2*(L>>4)+1}
// B (4x16 f32) layout: lane L holds col N=L%16, K-pair {2*(L>>4), 2*(L>>4)+1}
// C/D (16x16 f32):     acc[r] -> (M = r + 8*(L>>4), N = L%16)
// ---------------------------------------------------------------------------
__global__ __launch_bounds__(256) void gemm_kernel(const float* __restrict__ a1,
                                                   const float* __restrict__ a2,
                                                   const float* __restrict__ W,
                                                   const float* __restrict__ bias,
                                                   float*       __restrict__ h) {
  const int mTiles = N_NODES / 16;   // 3125 (exact)
  const int nTiles = H_DIM / 16;     // 4
  const int wave   = blockIdx.x * (256 / 32) + (threadIdx.x >> 5);
  if (wave >= mTiles * nTiles) return;   // wave-uniform: EXEC stays all-1s

  const int mTile = wave >> 2;
  const int nTile = wave & 3;
  const int lane  = threadIdx.x & 31;
  const int row   = lane & 15;           // M index (A) / N index (B,C,D)
  const int kh    = lane >> 4;           // which K half-pair this lane owns
  const int col   = nTile * 16 + row;

  v8f acc = {};
#pragma unroll
  for (int kb = 0; kb < 32; ++kb) {
    const int k = kb * 4 + kh * 2;       // lane's first K within this step
    // A row = concat(a1[node], a2[node]); contiguous pair -> float2 load
    const float* aptr = (kb < 16)
        ? (a1 + (long)(mTile * 16 + row) * H_DIM + k)
        : (a2 + (long)(mTile * 16 + row) * H_DIM + (k - 64));
    const float2 av = *(const float2*)aptr;
    v2f a; a.x = av.x; a.y = av.y;

    v2f b;                               // W is [128][64] row-major, 32 KB (L2-hot)
    b.x = W[(k + 0) * H_DIM + col];
    b.y = W[(k + 1) * H_DIM + col];

    acc = __builtin_amdgcn_wmma_f32_16x16x4_f32(
        /*neg_a=*/false, a, /*neg_b=*/false, b,
        /*c_mod=*/(short)0, acc, /*reuse_a=*/false, /*reuse_b=*/false);
  }

  const float bb = bias[col];
#pragma unroll
  for (int r = 0; r < 8; ++r) {
    const int node = mTile * 16 + r + kh * 8;
    float v = acc[r] + bb;
    v = v > 0.f ? v : 0.f;
    h[(long)node * H_DIM + col] += v;    // residual; unique (node,col) per lane
  }
}

// ---------------------------------------------------------------------------
// Launch: h -> d_out, then 2 layers of {zero, agg x2, wmma-gemm}.
// d_ws: a1 (N*H floats) | a2 (N*H floats)  = 25.6 MB
// ---------------------------------------------------------------------------
extern "C" void kernel_launch(void* const* d_in, const int* in_sizes, int n_in,
                              void* d_out, int out_size, void* d_ws, size_t ws_size,
                              hipStream_t stream) {
  const float* h     = (const float*)d_in[0];
  const float* feat1 = (const float*)d_in[1];
  const float* feat2 = (const float*)d_in[2];
  const float* W1    = (const float*)d_in[3];
  const float* b1    = (const float*)d_in[4];
  const float* W2    = (const float*)d_in[5];
  const float* b2    = (const float*)d_in[6];
  const int*   src1  = (const int*)d_in[7];
  const int*   dst1  = (const int*)d_in[8];
  const int*   src2  = (const int*)d_in[9];
  const int*   dst2  = (const int*)d_in[10];

  float* hbuf = (float*)d_out;                 // running node state, updated in place
  float* a1   = (float*)d_ws;
  float* a2   = a1 + (size_t)N_NODES * H_DIM;

  hipMemcpyAsync(hbuf, h, (size_t)N_NODES * H_DIM * sizeof(float),
                 hipMemcpyDeviceToDevice, stream);

  const int zero_n4     = (2 * N_NODES * H_DIM) / 4;          // 1.6M float4
  const int zero_blocks = (zero_n4 + 255) / 256;
  const int agg_blocks  = (E_EDGES * 32 + 255) / 256;         // 100000
  const int tiles       = (N_NODES / 16) * (H_DIM / 16);      // 12500 waves
  const int gemm_blocks = (tiles + 7) / 8;                    // 8 waves/block

  for (int layer = 0; layer < 2; ++layer) {
    const float* W = layer ? W2 : W1;
    const float* b = layer ? b2 : b1;

    zero_kernel<<<zero_blocks, 256, 0, stream>>>((float4*)a1, zero_n4);
    agg_kernel <<<agg_blocks, 256, 0, stream>>>(hbuf, feat1, src1, dst1, a1);
    agg_kernel <<<agg_blocks, 256, 0, stream>>>(hbuf, feat2, src2, dst2, a2);
    gemm_kernel<<<gemm_blocks, 256, 0, stream>>>(a1, a2, W, b, hbuf);
  }
}